// TGCN2_286
// MI455X (gfx1250) — compile-verified
//
#include <hip/hip_runtime.h>
#include <hip/hip_bf16.h>
#include <math.h>

typedef float v2f __attribute__((ext_vector_type(2)));
typedef float v8f __attribute__((ext_vector_type(8)));

// fp32 WMMA: D(16x16) = A(16x4) * B(4x16) + C ; 8-arg pattern per probe doc
#define WMMA_F32(a, b, c) \
  __builtin_amdgcn_wmma_f32_16x16x4_f32(false, (a), false, (b), (short)0, (c), false, false)

static constexpr int BB   = 4;
static constexpr int FIN  = 16;
static constexpr int FOUT = 32;

// ---------------- degree / normalization ----------------

__global__ void k_init_deg(float* deg, int n) {
  int i = blockIdx.x * blockDim.x + threadIdx.x;
  if (i < n) deg[i] = 1.0f;  // self-loop weight
}

__global__ void k_accum_deg(const long long* __restrict__ dst,
                            const float* __restrict__ w, float* deg, int e) {
  int i = blockIdx.x * blockDim.x + threadIdx.x;
  if (i < e) atomicAdd(&deg[dst[i]], w[i]);
}

__global__ void k_dinv(float* deg, int n) {
  int i = blockIdx.x * blockDim.x + threadIdx.x;
  if (i < n) {
    float d = deg[i];
    deg[i] = (d > 0.0f) ? rsqrtf(d) : 0.0f;
  }
}

// ---------------- XW = X @ [Wz | Wr | Wh]  ([M,16] x [16,96]) ----------------

__global__ void __launch_bounds__(256) k_xw_wmma(
    const float* __restrict__ X, const float* __restrict__ Wz,
    const float* __restrict__ Wr, const float* __restrict__ Wh,
    float* __restrict__ XW, int mtiles) {
  int wave = (int)((blockIdx.x * blockDim.x + threadIdx.x) >> 5);
  int lane = threadIdx.x & 31;
  if (wave >= mtiles) return;  // wave-uniform; EXEC stays all-1s for WMMA

  const float* Wg[3] = {Wz, Wr, Wh};
  int row0  = wave * 16;
  int m15   = lane & 15;
  int hi    = lane >> 4;
  int khalf = hi * 2;
  long long mrow = row0 + m15;

  // A fragments: lane<16 holds (K=4s, 4s+1), lane>=16 holds (K=4s+2, 4s+3) of row m15
  v2f a[4];
#pragma unroll
  for (int s = 0; s < 4; ++s) {
    int k0 = 4 * s + khalf;
    a[s].x = X[mrow * FIN + k0];
    a[s].y = X[mrow * FIN + k0 + 1];
  }

#pragma unroll
  for (int j = 0; j < 6; ++j) {  // 6 col-tiles of 16 -> cols 0..95
    const float* W = Wg[j >> 1];
    int col = (j & 1) * 16 + m15;
    v8f acc = {};
#pragma unroll
    for (int s = 0; s < 4; ++s) {
      int k0 = 4 * s + khalf;
      v2f b;
      b.x = W[k0 * FOUT + col];
      b.y = W[(k0 + 1) * FOUT + col];
      acc = WMMA_F32(a[s], b, acc);
    }
#pragma unroll
    for (int r = 0; r < 8; ++r) {
      long long orow = row0 + r + hi * 8;  // C/D layout: vgpr r -> M=r (lo half) / r+8 (hi half)
      XW[orow * 96 + j * 16 + m15] = acc[r];
    }
  }
}

// ---------------- C init (self loops) + edge scatter ----------------

__global__ void k_init_c(const float* __restrict__ XW, const float* __restrict__ dinv,
                         float* __restrict__ C, int n, long long total) {
  long long i = (long long)blockIdx.x * blockDim.x + threadIdx.x;
  if (i >= total) return;
  int row  = (int)(i / 96);
  int node = row % n;
  float di = dinv[node];
  C[i] = XW[i] * di * di;  // self-loop: norm = dinv[n]*1*dinv[n]
}

__global__ void k_scatter(const long long* __restrict__ src, const long long* __restrict__ dst,
                          const float* __restrict__ w, const float* __restrict__ dinv,
                          const float* __restrict__ XW, float* __restrict__ C,
                          int e, int n) {
  long long i = (long long)blockIdx.x * blockDim.x + threadIdx.x;
  if (i >= (long long)e * 96) return;
  int eid = (int)(i / 96);
  int f   = (int)(i % 96);
  long long s = src[eid];
  long long d = dst[eid];
  float nrm = dinv[s] * w[eid] * dinv[d];
#pragma unroll
  for (int b = 0; b < BB; ++b) {
    float msg = XW[((long long)b * n + s) * 96 + f] * nrm;
    atomicAdd(&C[((long long)b * n + d) * 96 + f], msg);
  }
}

// ---------------- fused GRU gate kernel ([M,64] x [64,32] x3 + pointwise) -----

__device__ __forceinline__ float sigmoidf(float x) { return 1.0f / (1.0f + expf(-x)); }

__global__ void __launch_bounds__(256) k_gates_wmma(
    const float* __restrict__ C, const float* __restrict__ H,
    const float* __restrict__ bz, const float* __restrict__ br, const float* __restrict__ bh,
    const float* __restrict__ Lz, const float* __restrict__ lbz,
    const float* __restrict__ Lr, const float* __restrict__ lbr,
    const float* __restrict__ Lh, const float* __restrict__ lbh,
    float* __restrict__ out, int mtiles) {
  __shared__ float rbuf[8][16][32];  // per-wave R tile (C-layout -> A-layout via LDS)
  int wave  = (int)((blockIdx.x * blockDim.x + threadIdx.x) >> 5);
  int wslot = threadIdx.x >> 5;
  int lane  = threadIdx.x & 31;
  if (wave >= mtiles) return;  // wave-uniform

  int row0  = wave * 16;
  int m15   = lane & 15;
  int hi    = lane >> 4;
  int khalf = hi * 2;
  long long mrow = row0 + m15;

  v8f zero = {};
  v8f accz[2] = {zero, zero};
  v8f accr[2] = {zero, zero};

  // K=64: first 32 = conv output (+conv bias), last 32 = H
  for (int s = 0; s < 16; ++s) {
    int k0 = 4 * s + khalf;
    v2f az, ar;
    if (s < 8) {
      az.x = C[mrow * 96 + k0] + bz[k0];
      az.y = C[mrow * 96 + k0 + 1] + bz[k0 + 1];
      ar.x = C[mrow * 96 + 32 + k0] + br[k0];
      ar.y = C[mrow * 96 + 32 + k0 + 1] + br[k0 + 1];
    } else {
      int kk = k0 - 32;
      float h0 = H[mrow * 32 + kk];
      float h1 = H[mrow * 32 + kk + 1];
      az.x = h0; az.y = h1;
      ar.x = h0; ar.y = h1;
    }
#pragma unroll
    for (int j = 0; j < 2; ++j) {
      int col = j * 16 + m15;
      v2f bzf, brf;
      bzf.x = Lz[k0 * 32 + col];
      bzf.y = Lz[(k0 + 1) * 32 + col];
      brf.x = Lr[k0 * 32 + col];
      brf.y = Lr[(k0 + 1) * 32 + col];
      accz[j] = WMMA_F32(az, bzf, accz[j]);
      accr[j] = WMMA_F32(ar, brf, accr[j]);
    }
  }

  // activate; keep Z in registers, spill R to per-wave LDS in row-major [16][32]
  v8f zf[2];
#pragma unroll
  for (int j = 0; j < 2; ++j) {
    int col = j * 16 + m15;
#pragma unroll
    for (int r = 0; r < 8; ++r) {
      zf[j][r] = sigmoidf(accz[j][r] + lbz[col]);
      rbuf[wslot][r + hi * 8][col] = sigmoidf(accr[j][r] + lbr[col]);
    }
  }

  // candidate gate: A = [ C_h + bh | H * R ]
  v8f acch[2] = {zero, zero};
  for (int s = 0; s < 16; ++s) {
    int k0 = 4 * s + khalf;
    v2f a;
    if (s < 8) {
      a.x = C[mrow * 96 + 64 + k0] + bh[k0];
      a.y = C[mrow * 96 + 64 + k0 + 1] + bh[k0 + 1];
    } else {
      int kk = k0 - 32;
      a.x = H[mrow * 32 + kk] * rbuf[wslot][m15][kk];
      a.y = H[mrow * 32 + kk + 1] * rbuf[wslot][m15][kk + 1];
    }
#pragma unroll
    for (int j = 0; j < 2; ++j) {
      int col = j * 16 + m15;
      v2f bf;
      bf.x = Lh[k0 * 32 + col];
      bf.y = Lh[(k0 + 1) * 32 + col];
      acch[j] = WMMA_F32(a, bf, acch[j]);
    }
  }

  // out = Z*H + (1-Z)*tanh(acch + lbh)
#pragma unroll
  for (int j = 0; j < 2; ++j) {
    int col = j * 16 + m15;
#pragma unroll
    for (int r = 0; r < 8; ++r) {
      long long orow = row0 + r + hi * 8;
      float ht = tanhf(acch[j][r] + lbh[col]);
      float hv = H[orow * 32 + col];
      float z  = zf[j][r];
      out[orow * 32 + col] = z * hv + (1.0f - z) * ht;
    }
  }
}

// ---------------- host launcher ----------------

extern "C" void kernel_launch(void* const* d_in, const int* in_sizes, int n_in,
                              void* d_out, int out_size, void* d_ws, size_t ws_size,
                              hipStream_t stream) {
  const float*     X   = (const float*)d_in[0];
  const long long* ei  = (const long long*)d_in[1];
  const float*     H   = (const float*)d_in[2];
  const float*     w   = (const float*)d_in[3];
  const float*     Wz  = (const float*)d_in[4];
  const float*     bz  = (const float*)d_in[5];
  const float*     Lz  = (const float*)d_in[6];
  const float*     lbz = (const float*)d_in[7];
  const float*     Wr  = (const float*)d_in[8];
  const float*     br  = (const float*)d_in[9];
  const float*     Lr  = (const float*)d_in[10];
  const float*     lbr = (const float*)d_in[11];
  const float*     Wh  = (const float*)d_in[12];
  const float*     bh  = (const float*)d_in[13];
  const float*     Lh  = (const float*)d_in[14];
  const float*     lbh = (const float*)d_in[15];
  (void)n_in; (void)ws_size;

  const int E  = in_sizes[3];                 // edge_weight count
  const int N  = in_sizes[2] / (BB * FOUT);   // from H = [B,N,32]
  const int M  = BB * N;                      // 200000, divisible by 16
  const int MT = (M + 15) / 16;

  char*  ws       = (char*)d_ws;
  size_t degBytes = (((size_t)N * sizeof(float)) + 255) / 256 * 256;
  size_t xwBytes  = (((size_t)M * 96 * sizeof(float)) + 255) / 256 * 256;
  float* deg = (float*)ws;                     // becomes dinv after k_dinv
  float* XW  = (float*)(ws + degBytes);
  float* Cg  = (float*)(ws + degBytes + xwBytes);

  const long long* srcI = ei;       // edge_index[0]
  const long long* dstI = ei + E;   // edge_index[1]

  k_init_deg<<<(N + 255) / 256, 256, 0, stream>>>(deg, N);
  k_accum_deg<<<(E + 255) / 256, 256, 0, stream>>>(dstI, w, deg, E);
  k_dinv<<<(N + 255) / 256, 256, 0, stream>>>(deg, N);

  k_xw_wmma<<<(MT + 7) / 8, 256, 0, stream>>>(X, Wz, Wr, Wh, XW, MT);

  long long totalC = (long long)M * 96;
  k_init_c<<<(int)((totalC + 255) / 256), 256, 0, stream>>>(XW, deg, Cg, N, totalC);

  long long totalE = (long long)E * 96;
  k_scatter<<<(int)((totalE + 255) / 256), 256, 0, stream>>>(srcI, dstI, w, deg, XW, Cg, E, N);

  k_gates_wmma<<<(MT + 7) / 8, 256, 0, stream>>>(Cg, H, bz, br, bh,
                                                 Lz, lbz, Lr, lbr, Lh, lbh,
                                                 (float*)d_out, MT);
}